// AdaptCNN_35974646071957
// MI455X (gfx1250) — compile-verified
//
#include <hip/hip_runtime.h>
#include <hip/hip_bf16.h>

// ---------------------------------------------------------------------------
// Fused "AdaptCNN" attention kernel for gfx1250 (MI455X), f32 end-to-end.
//
// Shapes: b=4, c=16, m=8, h=w=64, O=64.  out = [b,c,O,h,w] f32 (67 MB).
// Roofline: output stores (67 MB) dominate -> ~2.9 us at 23.3 TB/s; x (8.4 MB)
// is fully L2-resident even when re-read per (b,beta) pair. So: one fused
// kernel, nothing but `out` touches HBM.
//
// Raw-view identity (B = b*64 + beta):
//   uq[B,i,p] = s * sum_m w_q[beta,m] x[b,i,m,pix]
//   uk[B,j,p] =     sum_m w_k[16*(beta&3)+j,m] x[b,beta>>2,m,pix] + 2*pe(pix)
//   uv[B,j,p] =     sum_m w_v[16*(beta&3)+j,m] x[b,beta>>2,m,pix]
//   att = softmax_j(uq . uk);  out[B,i,p] = sum_j att[i,j] uv[j,p]
//   final[b,i,beta,y,x] = fold(out) + mean_m x[b,i,m,y,x]
//
// gfx1250 features used:
//   * TENSOR_LOAD_TO_LDS (TDM, 3D descriptor) stages the x tile; wave 0
//     issues, s_wait_tensorcnt + workgroup barrier publishes it.
//   * V_WMMA_F32_16X16X4_F32 for k/v projections (K=8 as 2 chained WMMAs)
//     and for the per-patch 16x16 (K=4) attention matmul (computed as att^T
//     so softmax-over-j is 8 VGPRs + one lane^16 shuffle in wave32).
// ---------------------------------------------------------------------------

typedef float        v2f __attribute__((ext_vector_type(2)));
typedef float        v8f __attribute__((ext_vector_type(8)));
typedef unsigned int v4u __attribute__((ext_vector_type(4)));
typedef int          v4i __attribute__((ext_vector_type(4)));
typedef int          v8i __attribute__((ext_vector_type(8)));

#define B_      4
#define C_      16
#define M_      8
#define HW_     64
#define O_      64
#define SCALING 0.125f           // h^-0.5 = 64^-0.5
#define LSTEP   (2.0f / 63.0f)   // linspace(-1,1,64) step

__global__ __launch_bounds__(256)
void adapt_attn_fused(const float* __restrict__ x,
                      const float* __restrict__ w_q,
                      const float* __restrict__ w_k,
                      const float* __restrict__ w_v,
                      const float* __restrict__ w_p,
                      float* __restrict__ out)
{
    extern __shared__ float smem[];
    float* xs  = smem;                    // [C_][M_][256] (TDM fill order) = 32768 f
    float* wsc = smem + C_ * M_ * 256;    // per-wave scratch: 8 waves * 768 f

    const int bi   = blockIdx.x >> 6;     // batch
    const int beta = blockIdx.x & 63;     // q output channel / attention block
    const int y0   = blockIdx.y * 4;      // first pixel row of this tile

    const int cj    = beta >> 2;          // x channel feeding k/v
    const int jbase = (beta & 3) * 16;    // first w_k/w_v row for this block

    const int tid  = threadIdx.x;
    const int lane = tid & 31;
    const int wave = tid >> 5;

    // ---------------- stage x[bi, :, :, y0:y0+4, :] into LDS ----------------
    // 3D tile: dim0 = 256 contiguous floats (4 rows x 64), dim1 = m (stride
    // 4096 elems), dim2 = ci (stride 32768 elems) -> xs[((ci*8+m)*256)+pix].
#if __has_builtin(__builtin_amdgcn_tensor_load_to_lds)
    if (wave == 0) {
        const unsigned long long gaddr =
            (unsigned long long)(const void*)(x + ((size_t)bi * C_ * M_) * HW_ * HW_
                                                + (size_t)y0 * HW_);
        v4u g0;
        g0[0] = 1u;                                   // count=1 (valid descriptor)
        g0[1] = 0u;                                   // lds_addr = dynamic-LDS base
        g0[2] = (unsigned)(gaddr & 0xFFFFFFFFu);      // global_addr[31:0]
        g0[3] = (unsigned)((gaddr >> 32) & 0x01FFFFFFu) | (2u << 30); // [56:32]|type=2
        v8i g1;
        g1[0] = (int)(2u << 16);                      // data_size=2 (4 bytes)
        g1[1] = (int)(256u << 16);                    // tensor_dim0[15:0]=256
        g1[2] = (int)(8u << 16);                      // tensor_dim1[15:0]=8
        g1[3] = (int)(256u << 16);                    // tile_dim0=256
        g1[4] = (int)(8u | (16u << 16));              // tile_dim1=8, tile_dim2=16
        g1[5] = 4096;                                 // tensor_dim0_stride[31:0]
        g1[6] = (int)(32768u << 16);                  // tensor_dim1_stride[15:0]=32768
        g1[7] = 0;                                    // tensor_dim1_stride[47:16]=0
        v4i g2; g2[0] = 16; g2[1] = 0; g2[2] = 0; g2[3] = 0;   // tensor_dim2=16
        v4i g3; g3[0] = 0;  g3[1] = 0; g3[2] = 0; g3[3] = 0;
#if defined(__clang_major__) && (__clang_major__ >= 23)
        v8i g4; g4[0]=0; g4[1]=0; g4[2]=0; g4[3]=0; g4[4]=0; g4[5]=0; g4[6]=0; g4[7]=0;
        __builtin_amdgcn_tensor_load_to_lds(g0, g1, g2, g3, g4, 0);
#else
        __builtin_amdgcn_tensor_load_to_lds(g0, g1, g2, g3, 0);
#endif
#if __has_builtin(__builtin_amdgcn_s_wait_tensorcnt)
        __builtin_amdgcn_s_wait_tensorcnt(0);
#else
        asm volatile("s_wait_tensorcnt 0x0" ::: "memory");
#endif
    }
#else
    // Fallback: coalesced float4 staging (same LDS layout as the TDM path).
    #pragma unroll
    for (int it = 0; it < 32; ++it) {
        const int lin4 = it * 256 + tid;              // float4 index over (ci,m,pix/4)
        const int ci  = lin4 >> 9;
        const int rem = lin4 & 511;
        const int m   = rem >> 6;
        const int p4  = rem & 63;
        const float4 v = *(const float4*)(x + (((size_t)(bi * C_ + ci) * M_ + m) * HW_
                                               + y0) * HW_ + p4 * 4);
        ((float4*)xs)[lin4] = v;
    }
#endif
    __syncthreads();

    // ---------------- wave-uniform weights ---------------------------------
    float wqr[M_];
    #pragma unroll
    for (int m = 0; m < M_; ++m) wqr[m] = w_q[beta * M_ + m];
    const float wp0 = w_p[0], wp1 = w_p[1];

    // A fragments for k/v WMMAs (16x4 f32 A layout):
    //   VGPR0: K = kb (lanes 0-15 kb=0, 16-31 kb=2), VGPR1: K = kb+1, M = lane&15
    const int jA = lane & 15;
    const int kb = (lane < 16) ? 0 : 2;
    v2f akl, akh, avl, avh;
    {
        const int base = (jbase + jA) * M_ + kb;
        akl.x = w_k[base];     akl.y = w_k[base + 1];
        akh.x = w_k[base + 4]; akh.y = w_k[base + 5];
        avl.x = w_v[base];     avl.y = w_v[base + 1];
        avh.x = w_v[base + 4]; avh.y = w_v[base + 5];
    }

    // wave -> 8 patches: 2 patch rows x 32 cols in this tile
    const int prow = wave >> 2;            // patch row within tile (0/1)
    const int pcb  = (wave & 3) * 8;       // patch col base
    float* ks = wsc + wave * 768;          // [16 j][16 pix]
    float* qs = ks + 256;                  // [16 pix][16 i]
    float* vs = qs + 256;                  // [16 j][16 pix]

    const int joff = (lane < 16) ? 0 : 8;  // which half of att rows this lane holds
    const int ph   = lane >> 4;            // row half (also fold dy at the end)

    for (int g = 0; g < 2; ++g) {
        const int pc0 = pcb + g * 4;       // first patch col of this 4-patch group

        // ---- B fragments (x[cj] at this lane's pixel), pixel N = lane&15 ---
        const int pixL = lane & 15;
        const int ttL  = pixL >> 2, ppL = pixL & 3;
        const int yyL  = prow * 2 + (ppL >> 1);
        const int xcL  = (pc0 + ttL) * 2 + (ppL & 1);
        const int pIdx = yyL * 64 + xcL;
        v2f bl, bh;
        bl.x = xs[(cj * M_ + kb)     * 256 + pIdx];
        bl.y = xs[(cj * M_ + kb + 1) * 256 + pIdx];
        bh.x = xs[(cj * M_ + kb + 4) * 256 + pIdx];
        bh.y = xs[(cj * M_ + kb + 5) * 256 + pIdx];

        // ---- k and v projections: two chained 16x16x4 f32 WMMAs each ------
        v8f kacc = {0.f, 0.f, 0.f, 0.f, 0.f, 0.f, 0.f, 0.f};
        kacc = __builtin_amdgcn_wmma_f32_16x16x4_f32(false, akl, false, bl,
                                                     (short)0, kacc, false, false);
        kacc = __builtin_amdgcn_wmma_f32_16x16x4_f32(false, akh, false, bh,
                                                     (short)0, kacc, false, false);
        v8f vacc = {0.f, 0.f, 0.f, 0.f, 0.f, 0.f, 0.f, 0.f};
        vacc = __builtin_amdgcn_wmma_f32_16x16x4_f32(false, avl, false, bl,
                                                     (short)0, vacc, false, false);
        vacc = __builtin_amdgcn_wmma_f32_16x16x4_f32(false, avh, false, bh,
                                                     (short)0, vacc, false, false);

        // uk += 2*pe(pixel); D layout: vgpr rr -> j = rr + joff, N = pixL
        const float pe2 = 2.0f * (wp0 * (-1.0f + LSTEP * (float)xcL) +
                                  wp1 * (-1.0f + LSTEP * (float)(y0 + yyL)));
        #pragma unroll
        for (int rr = 0; rr < 8; ++rr) {
            ks[(rr + joff) * 16 + pixL] = kacc[rr] + pe2;
            vs[(rr + joff) * 16 + pixL] = vacc[rr];
        }

        // ---- q matvec + residual mean, row-based (lane = (i, row half)) ---
        // Lane owns i = lane&15 and the 8 contiguous pixels of row prow*2+ph.
        const int iQ      = lane & 15;
        const int yyq     = prow * 2 + ph;
        const int pixBase = yyq * 64 + pc0 * 2;
        float qr[8] = {0,0,0,0,0,0,0,0};
        float rs[8] = {0,0,0,0,0,0,0,0};
        #pragma unroll
        for (int m = 0; m < M_; ++m) {
            const float4* xr = (const float4*)&xs[(iQ * M_ + m) * 256 + pixBase];
            const float4 a = xr[0], b = xr[1];
            qr[0] += wqr[m] * a.x; qr[1] += wqr[m] * a.y;
            qr[2] += wqr[m] * a.z; qr[3] += wqr[m] * a.w;
            qr[4] += wqr[m] * b.x; qr[5] += wqr[m] * b.y;
            qr[6] += wqr[m] * b.z; qr[7] += wqr[m] * b.w;
            rs[0] += a.x; rs[1] += a.y; rs[2] += a.z; rs[3] += a.w;
            rs[4] += b.x; rs[5] += b.y; rs[6] += b.z; rs[7] += b.w;
        }
        #pragma unroll
        for (int xo = 0; xo < 8; ++xo) {
            const int pixq = (xo >> 1) * 4 + ph * 2 + (xo & 1);
            qs[pixq * 16 + iQ] = qr[xo] * SCALING;
        }

        // ---- attention per patch ------------------------------------------
        #pragma unroll
        for (int tt = 0; tt < 4; ++tt) {
            // A = uk[j, p]: p-pair {kb,kb+1} from ks (b64), M = j = lane&15
            v2f aA;
            {
                const float2 af = *(const float2*)&ks[(lane & 15) * 16 + tt * 4 + kb];
                aA.x = af.x; aA.y = af.y;
            }
            // B = uq^T[p, i]: VGPR0 -> p=kb, VGPR1 -> p=kb+1, N = i = lane&15
            v2f bB;
            bB.x = qs[(tt * 4 + kb)     * 16 + (lane & 15)];
            bB.y = qs[(tt * 4 + kb + 1) * 16 + (lane & 15)];

            v8f att = {0.f, 0.f, 0.f, 0.f, 0.f, 0.f, 0.f, 0.f};
            att = __builtin_amdgcn_wmma_f32_16x16x4_f32(false, aA, false, bB,
                                                        (short)0, att, false, false);
            // att^T[j = rr+joff][i = lane&15] now lives in att[rr].

            // softmax over j: 8 local rows + 8 rows in lane^16
            float mx = att[0];
            #pragma unroll
            for (int rr = 1; rr < 8; ++rr) mx = fmaxf(mx, att[rr]);
            mx = fmaxf(mx, __shfl_xor(mx, 16, 32));
            float e[8], s = 0.0f;
            #pragma unroll
            for (int rr = 0; rr < 8; ++rr) { e[rr] = __expf(att[rr] - mx); s += e[rr]; }
            s += __shfl_xor(s, 16, 32);
            const float inv = __builtin_amdgcn_rcpf(s);

            // out[i,p] = sum_j att_n[j,i] * uv[j,p]; partial over this lane's j
            float po0 = 0.f, po1 = 0.f, po2 = 0.f, po3 = 0.f;
            #pragma unroll
            for (int rr = 0; rr < 8; ++rr) {
                const float a = e[rr] * inv;
                const float4 vv = *(const float4*)&vs[(rr + joff) * 16 + tt * 4];
                po0 += a * vv.x; po1 += a * vv.y; po2 += a * vv.z; po3 += a * vv.w;
            }
            po0 += __shfl_xor(po0, 16, 32);
            po1 += __shfl_xor(po1, 16, 32);
            po2 += __shfl_xor(po2, 16, 32);
            po3 += __shfl_xor(po3, 16, 32);

            // fold + residual (from rs[], same row as this lane's q row) + store
            const int iO = lane & 15;
            const int yg = y0 + prow * 2 + ph;
            const int xg = (pc0 + tt) * 2;
            const float res0 = rs[tt * 2]     * 0.125f;
            const float res1 = rs[tt * 2 + 1] * 0.125f;
            float2 o2;
            o2.x = ((ph == 0) ? po0 : po2) + res0;
            o2.y = ((ph == 0) ? po1 : po3) + res1;
            float* op = out + ((((size_t)bi * C_ + iO) * O_ + beta) * HW_ + yg) * HW_ + xg;
            *(float2*)op = o2;
        }
    }
}

extern "C" void kernel_launch(void* const* d_in, const int* in_sizes, int n_in,
                              void* d_out, int out_size, void* d_ws, size_t ws_size,
                              hipStream_t stream) {
    (void)in_sizes; (void)n_in; (void)out_size; (void)d_ws; (void)ws_size;
    const float* x  = (const float*)d_in[0];
    const float* wq = (const float*)d_in[1];
    const float* wk = (const float*)d_in[2];
    const float* wv = (const float*)d_in[3];
    const float* wp = (const float*)d_in[4];
    float* out = (float*)d_out;

    dim3 grid(B_ * O_, 16, 1);   // (b,beta) x 4-row tiles
    dim3 block(256, 1, 1);       // 8 wave32
    const size_t shmem = (size_t)(C_ * M_ * 256 + 8 * 768) * sizeof(float); // 152 KB
    adapt_attn_fused<<<grid, block, shmem, stream>>>(x, wq, wk, wv, wp, out);
}